// APPNP_net_84954453115011
// MI455X (gfx1250) — compile-verified
//
#include <hip/hip_runtime.h>
#include <stdint.h>

// APPNP (K=2, alpha=0.5) + fused ReLU + Linear for MI455X (gfx1250, wave32).
// Phase 1: build CSR by dst (int atomics + LDS scan), degree via float atomics.
// Phase 2: per-hop SpMM, one wave32 per destination row, no float atomics.
// Phase 3: out = relu(h) @ W^T + b via V_WMMA_F32_16X16X4_F32 (exact fp32).

#define ALPHA_F 0.5f
#define OMA_F   0.5f   // 1 - alpha
#define DFEAT   256

typedef __attribute__((ext_vector_type(2))) float v2f;
typedef __attribute__((ext_vector_type(8))) float v8f;

// ---------------- CSR construction ----------------

__global__ void init_kernel(float* __restrict__ deg, unsigned* __restrict__ count, int N) {
  int i = blockIdx.x * blockDim.x + threadIdx.x;
  if (i < N) { deg[i] = 1.0f; count[i] = 0u; }   // self-loop weight 1 folded into deg
}

__global__ void deg_count_kernel(const long long* __restrict__ ei, const float* __restrict__ w,
                                 float* __restrict__ deg, unsigned* __restrict__ count, int E) {
  int e = blockIdx.x * blockDim.x + threadIdx.x;
  if (e < E) {
    int dst = (int)ei[(long long)E + e];
    atomicAdd(&deg[dst], w[e]);
    atomicAdd(&count[dst], 1u);
  }
}

__global__ void dinv_kernel(const float* __restrict__ deg, float* __restrict__ dinv, int N) {
  int i = blockIdx.x * blockDim.x + threadIdx.x;
  if (i < N) {
    float d = deg[i];
    dinv[i] = d > 0.0f ? rsqrtf(fmaxf(d, 1e-12f)) : 0.0f;
  }
}

// Single-workgroup chunked Hillis-Steele exclusive scan (N=100k -> ~98 chunks; one-time cost).
__global__ void scan_kernel(const unsigned* __restrict__ count,
                            unsigned* __restrict__ rowptr, unsigned* __restrict__ cursor, int N) {
  __shared__ unsigned sh[1024];
  __shared__ unsigned carry;
  int t = threadIdx.x;
  if (t == 0) carry = 0u;
  __syncthreads();
  for (int base = 0; base < N; base += 1024) {
    int i = base + t;
    unsigned v = (i < N) ? count[i] : 0u;
    sh[t] = v;
    __syncthreads();
    for (int off = 1; off < 1024; off <<= 1) {
      unsigned add = (t >= off) ? sh[t - off] : 0u;
      __syncthreads();
      sh[t] += add;
      __syncthreads();
    }
    unsigned incl = sh[t];
    if (i < N) {
      unsigned ex = carry + (incl - v);
      rowptr[i] = ex;
      cursor[i] = ex;
    }
    __syncthreads();
    if (t == 1023) carry += incl;
    __syncthreads();
  }
}

__global__ void fill_kernel(const long long* __restrict__ ei, const float* __restrict__ w,
                            const float* __restrict__ dinv, unsigned* __restrict__ cursor,
                            unsigned* __restrict__ col, float* __restrict__ val, int E) {
  int e = blockIdx.x * blockDim.x + threadIdx.x;
  if (e < E) {
    int s = (int)ei[e];
    int d = (int)ei[(long long)E + e];
    unsigned pos = atomicAdd(&cursor[d], 1u);
    col[pos] = (unsigned)s;
    val[pos] = dinv[s] * w[e] * dinv[d];
  }
}

// ---------------- Propagation hop: one wave32 per destination row ----------------
// lane holds 8 consecutive floats (2x float4) of the 256-wide row; gathers hit L2.

__global__ __launch_bounds__(256) void hop_kernel(
    const float* __restrict__ hin, const float* __restrict__ x, float* __restrict__ hout,
    const unsigned* __restrict__ rowptr, const unsigned* __restrict__ rowend,
    const unsigned* __restrict__ col, const float* __restrict__ val,
    const float* __restrict__ dinv, int N) {
  int wave = threadIdx.x >> 5;
  int lane = threadIdx.x & 31;
  int node = blockIdx.x * 8 + wave;
  if (node >= N) return;                       // wave-uniform

  const float4* hin4 = (const float4*)hin;
  int fb = lane * 2;                           // float4 index within the 64-float4 row
  float4 a0 = make_float4(0.f, 0.f, 0.f, 0.f);
  float4 a1 = make_float4(0.f, 0.f, 0.f, 0.f);

  unsigned p0 = rowptr[node], p1 = rowend[node];
  long long rowBase;
  for (unsigned p = p0; p < p1; ++p) {
    unsigned c = col[p];
    float v = val[p];
    rowBase = (long long)c * 64;
    float4 ha = hin4[rowBase + fb];
    float4 hb = hin4[rowBase + fb + 1];
    a0.x = fmaf(v, ha.x, a0.x); a0.y = fmaf(v, ha.y, a0.y);
    a0.z = fmaf(v, ha.z, a0.z); a0.w = fmaf(v, ha.w, a0.w);
    a1.x = fmaf(v, hb.x, a1.x); a1.y = fmaf(v, hb.y, a1.y);
    a1.z = fmaf(v, hb.z, a1.z); a1.w = fmaf(v, hb.w, a1.w);
  }
  // self-loop term: dinv[i]^2 * hin[i]
  float di = dinv[node];
  float vs = di * di;
  rowBase = (long long)node * 64;
  float4 hs0 = hin4[rowBase + fb];
  float4 hs1 = hin4[rowBase + fb + 1];
  a0.x = fmaf(vs, hs0.x, a0.x); a0.y = fmaf(vs, hs0.y, a0.y);
  a0.z = fmaf(vs, hs0.z, a0.z); a0.w = fmaf(vs, hs0.w, a0.w);
  a1.x = fmaf(vs, hs1.x, a1.x); a1.y = fmaf(vs, hs1.y, a1.y);
  a1.z = fmaf(vs, hs1.z, a1.z); a1.w = fmaf(vs, hs1.w, a1.w);

  const float4* x4 = (const float4*)x;
  float4 x0 = x4[rowBase + fb];
  float4 x1 = x4[rowBase + fb + 1];
  float4 o0, o1;
  o0.x = OMA_F * a0.x + ALPHA_F * x0.x; o0.y = OMA_F * a0.y + ALPHA_F * x0.y;
  o0.z = OMA_F * a0.z + ALPHA_F * x0.z; o0.w = OMA_F * a0.w + ALPHA_F * x0.w;
  o1.x = OMA_F * a1.x + ALPHA_F * x1.x; o1.y = OMA_F * a1.y + ALPHA_F * x1.y;
  o1.z = OMA_F * a1.z + ALPHA_F * x1.z; o1.w = OMA_F * a1.w + ALPHA_F * x1.w;
  float4* hout4 = (float4*)hout;
  hout4[rowBase + fb] = o0;
  hout4[rowBase + fb + 1] = o1;
}

// ---------------- Dense epilogue: out = relu(h) @ W^T + b via fp32 WMMA ----------------
// Wave computes 16 rows x 64 cols. K loop steps by 4 (V_WMMA_F32_16X16X4_F32).
// A(16x4): lane holds M = lane&15, K-pair = 2*(lane>>4). B(4x16): lane holds N = lane&15,
// same K-pair. C/D(16x16): VGPR v -> row v + 8*(lane>>4), col = lane&15 (ISA 7.12.2).

__global__ __launch_bounds__(256) void gemm_kernel(
    const float* __restrict__ h, const float* __restrict__ W,
    const float* __restrict__ bias, float* __restrict__ out, int Ntiles) {
  int wave = threadIdx.x >> 5;
  int lane = threadIdx.x & 31;
  int rt = blockIdx.x * 8 + wave;
  if (rt >= Ntiles) return;                    // wave-uniform; EXEC stays all-ones
  int r0 = rt * 16;
  int c0 = blockIdx.y * 64;
  int m = lane & 15;
  int khalf = lane >> 4;                       // 0 or 1

  v8f acc[4];
#pragma unroll
  for (int t = 0; t < 4; ++t) {
    float bv = bias[c0 + t * 16 + m];
#pragma unroll
    for (int j = 0; j < 8; ++j) acc[t][j] = bv;
  }

  const v2f* hrow = (const v2f*)(h + (long long)(r0 + m) * DFEAT);
  const v2f* wrow0 = (const v2f*)(W + (long long)(c0 + 0 * 16 + m) * DFEAT);
  const v2f* wrow1 = (const v2f*)(W + (long long)(c0 + 1 * 16 + m) * DFEAT);
  const v2f* wrow2 = (const v2f*)(W + (long long)(c0 + 2 * 16 + m) * DFEAT);
  const v2f* wrow3 = (const v2f*)(W + (long long)(c0 + 3 * 16 + m) * DFEAT);

#pragma unroll 4
  for (int k0 = 0; k0 < DFEAT; k0 += 4) {
    int kp = (k0 >> 1) + khalf;                // v2f index: K-pair for this lane half
    v2f a = hrow[kp];
    a.x = fmaxf(a.x, 0.0f);                    // fused ReLU
    a.y = fmaxf(a.y, 0.0f);
    v2f b0 = wrow0[kp], b1 = wrow1[kp], b2 = wrow2[kp], b3 = wrow3[kp];
    acc[0] = __builtin_amdgcn_wmma_f32_16x16x4_f32(false, a, false, b0, (short)0, acc[0], false, false);
    acc[1] = __builtin_amdgcn_wmma_f32_16x16x4_f32(false, a, false, b1, (short)0, acc[1], false, false);
    acc[2] = __builtin_amdgcn_wmma_f32_16x16x4_f32(false, a, false, b2, (short)0, acc[2], false, false);
    acc[3] = __builtin_amdgcn_wmma_f32_16x16x4_f32(false, a, false, b3, (short)0, acc[3], false, false);
  }

  int rbase = r0 + (khalf << 3);
#pragma unroll
  for (int t = 0; t < 4; ++t) {
    int colo = c0 + t * 16 + m;
#pragma unroll
    for (int v = 0; v < 8; ++v)
      out[(long long)(rbase + v) * DFEAT + colo] = acc[t][v];
  }
}

// ---------------- host-side launch ----------------

extern "C" void kernel_launch(void* const* d_in, const int* in_sizes, int n_in,
                              void* d_out, int out_size, void* d_ws, size_t ws_size,
                              hipStream_t stream) {
  const float*     x  = (const float*)d_in[0];
  const long long* ei = (const long long*)d_in[1];   // int64 edge_index [2,E]
  const float*     w  = (const float*)d_in[2];
  const float*     Wm = (const float*)d_in[3];
  const float*     b  = (const float*)d_in[4];

  const int N = in_sizes[0] / DFEAT;
  const int E = in_sizes[2];

  // workspace carve-up (256B aligned)
  char* p = (char*)d_ws;
  auto carve = [&](size_t bytes) -> void* {
    void* r = (void*)p;
    p += (bytes + 255) & ~(size_t)255;
    return r;
  };
  float*    deg    = (float*)   carve((size_t)N * 4);
  float*    dinv   = (float*)   carve((size_t)N * 4);
  unsigned* count  = (unsigned*)carve((size_t)N * 4);
  unsigned* rowptr = (unsigned*)carve((size_t)N * 4);
  unsigned* cursor = (unsigned*)carve((size_t)N * 4);
  unsigned* col    = (unsigned*)carve((size_t)E * 4);
  float*    val    = (float*)   carve((size_t)E * 4);
  float*    h2     = (float*)   carve((size_t)N * DFEAT * 4);
  float*    h1     = (float*)d_out;                  // reuse output buffer for hop-1 result
  float*    outp   = (float*)d_out;

  dim3 blk(256);
  dim3 gN((N + 255) / 256), gE((E + 255) / 256);

  init_kernel<<<gN, blk, 0, stream>>>(deg, count, N);
  deg_count_kernel<<<gE, blk, 0, stream>>>(ei, w, deg, count, E);
  dinv_kernel<<<gN, blk, 0, stream>>>(deg, dinv, N);
  scan_kernel<<<1, 1024, 0, stream>>>(count, rowptr, cursor, N);
  fill_kernel<<<gE, blk, 0, stream>>>(ei, w, dinv, cursor, col, val, E);
  // after fill, cursor[i] == row end

  dim3 gHop((N + 7) / 8);
  hop_kernel<<<gHop, blk, 0, stream>>>(x,  x, h1, rowptr, cursor, col, val, dinv, N);
  hop_kernel<<<gHop, blk, 0, stream>>>(h1, x, h2, rowptr, cursor, col, val, dinv, N);

  int Ntiles = N / 16;                               // 100000/16 = 6250
  dim3 gGemm((Ntiles + 7) / 8, DFEAT / 64);
  gemm_kernel<<<gGemm, blk, 0, stream>>>(h2, Wm, b, outp, Ntiles);
}